// MentionScoresHead_1494648619663
// MI455X (gfx1250) — compile-verified
//
#include <hip/hip_runtime.h>
#include <hip/hip_bf16.h>

// Problem constants
#define BB   32
#define LL   512
#define DD   1024
#define TT   256
#define KOUT 136393            // number of kept (s,e) pairs per batch (e - s <= 10)

typedef float v2f __attribute__((ext_vector_type(2)));
typedef float v8f __attribute__((ext_vector_type(8)));

#define WROW 1032              // padded LDS row stride (dwords): 8-byte aligned,
                               // 1032 % 64 == 8 -> rows 0..3 land on distinct banks

__device__ __forceinline__ float neg_inf() { return -__builtin_inff(); }

// ---------------------------------------------------------------------------
// Kernel 1: logits = text_encodings @ W via V_WMMA_F32_16X16X4_F32.
// One wave per 16-token row tile.  W is staged in LDS *transposed*:
// smWt[n][k] for n=0..2, plus a dummy all-zero row n=3.  Lane l reads row
// min(l&15, 3): lanes 3..15 broadcast-read the zero row, so the B-fragment
// load is a single unconditional ds_load_b64 (no exec-mask divergence).
// Grid: 128 blocks x 256 threads (8 waves) -> 1024 row tiles.
// ---------------------------------------------------------------------------
__global__ __launch_bounds__(256) void mention_logits_wmma(
    const float* __restrict__ text,     // (B*L, D)
    const float* __restrict__ Wg,       // (D, 3)
    float* __restrict__ logits)         // (B*L, 3)
{
    __shared__ float smWt[4][WROW];     // ~16.5 KB

    for (int i = threadIdx.x; i < DD; i += 256) {
        const float w0 = Wg[i * 3 + 0];
        const float w1 = Wg[i * 3 + 1];
        const float w2 = Wg[i * 3 + 2];
        smWt[0][i] = w0;
        smWt[1][i] = w1;
        smWt[2][i] = w2;
        smWt[3][i] = 0.0f;
    }
    __syncthreads();

    const int wave = threadIdx.x >> 5;            // 0..7
    const int lane = threadIdx.x & 31;
    const int tile = blockIdx.x * 8 + wave;       // 0..1023
    const int half = lane >> 4;                   // 0 or 1
    const int l15  = lane & 15;
    const int wrow = (l15 < 3) ? l15 : 3;         // B-fragment column (3 = zeros)

    // A fragment: row = tile*16 + l15, contiguous K pair at k0 + half*2
    const float* arow = text + (size_t)(tile * 16 + l15) * DD + half * 2;
    // B fragment: W column wrow, same K pair (8-byte aligned: half*2 even)
    const float* brow = &smWt[wrow][half * 2];

    v8f acc = {0.f, 0.f, 0.f, 0.f, 0.f, 0.f, 0.f, 0.f};

    #pragma unroll 8
    for (int k0 = 0; k0 < DD; k0 += 4) {
        const v2f a  = *(const v2f*)(arow + k0);   // global_load_b64
        const v2f bm = *(const v2f*)(brow + k0);   // ds_load_b64 (uncond.)
        acc = __builtin_amdgcn_wmma_f32_16x16x4_f32(
            false, a, false, bm, (short)0, acc, false, false);
    }

    // C/D layout: lane l, vgpr v -> row (v + 8*half), col l15
    if (l15 < 3) {
        #pragma unroll
        for (int v = 0; v < 8; ++v) {
            const int row = tile * 16 + v + half * 8;
            logits[(size_t)row * 3 + l15] = acc[v];
        }
    }
}

// ---------------------------------------------------------------------------
// Kernel 2: per batch — bias, masking, cumsum of mention_lp, token-start/end
// scatter masks; fold into startc[s] = start - csum_prev[s] (or -inf) and
// endc[e] = end + csum[e] (or -inf).  Grid: 32 blocks x 512 threads.
// ---------------------------------------------------------------------------
__global__ __launch_bounds__(512) void mention_fold(
    const float* __restrict__ logits,   // (B*L, 3)
    const int*   __restrict__ mask,     // (B, L)
    const int*   __restrict__ tmap,     // (B, T, 2)
    const float* __restrict__ bias,     // (3,)
    float* __restrict__ startc,         // (B, L)
    float* __restrict__ endc)           // (B, L)
{
    __shared__ float sm[LL];
    __shared__ int tsA[LL];
    __shared__ int teA[LL];

    const int b = blockIdx.x;
    const int t = threadIdx.x;

    const float b0 = bias[0], b1 = bias[1], b2 = bias[2];
    const size_t idx = (size_t)b * LL + t;

    const float l0 = logits[idx * 3 + 0];
    const float l1 = logits[idx * 3 + 1];
    const float l2 = logits[idx * 3 + 2];
    const bool vtok = (mask[idx] == 1);

    const float start_lp   = vtok ? (l0 + b0) : neg_inf();
    const float end_lp     = vtok ? (l1 + b1) : neg_inf();
    const float mention_lp = vtok ? (l2 + b2) : neg_inf();

    // token-start / token-end scatter masks
    tsA[t] = 0;
    teA[t] = 0;
    __syncthreads();
    if (t < TT) {
        const int s0 = tmap[((size_t)b * TT + t) * 2 + 0];      // in [1, L)
        const int e1 = tmap[((size_t)b * TT + t) * 2 + 1] - 1;  // in [0, L-1)
        tsA[s0] = 1;
        teA[e1] = 1;
    }
    __syncthreads();
    if (t == 0) { tsA[0] = 0; teA[0] = 0; }

    // inclusive scan of mention_lp over L=512 (Hillis-Steele in LDS)
    sm[t] = mention_lp;
    __syncthreads();
    #pragma unroll
    for (int off = 1; off < LL; off <<= 1) {
        const float v = (t >= off) ? sm[t - off] : 0.0f;
        __syncthreads();
        sm[t] += v;
        __syncthreads();
    }
    const float csum  = sm[t];
    const float cprev = (t > 0) ? sm[t - 1] : 0.0f;

    startc[idx] = tsA[t] ? (start_lp - cprev) : neg_inf();
    endc[idx]   = teA[t] ? (end_lp + csum)    : neg_inf();
}

// ---------------------------------------------------------------------------
// Kernel 3: stream out kept pairs.  Block = (s-row, batch); entries for row s
// are e in [0, min(s+10, 511)], written at closed-form offset off(s).
// Scores: coalesced b32 stores.  Bounds: one b64 store per pair.
// ---------------------------------------------------------------------------
__global__ __launch_bounds__(256) void mention_emit(
    const float* __restrict__ startc,   // (B, L)
    const float* __restrict__ endc,     // (B, L)
    float* __restrict__ out_scores,     // (B, KOUT)
    int*   __restrict__ out_bounds)     // (B, KOUT, 2)
{
    const int s = blockIdx.x;           // 0..511
    const int b = blockIdx.y;           // 0..31

    const int n_e = (s + 11 < LL) ? (s + 11) : LL;
    const int off = (s <= 501) ? (s * (s + 21)) / 2
                               : 131273 + (s - 502) * LL;

    const float sc_s = startc[(size_t)b * LL + s];
    float* srow = out_scores + (size_t)b * KOUT + off;
    unsigned long long* brow =
        (unsigned long long*)(out_bounds + ((size_t)b * KOUT + off) * 2);

    const unsigned long long s_lo = (unsigned)s;   // low word = bounds[...,0]

    for (int e = threadIdx.x; e < n_e; e += 256) {
        const float v = (e >= s) ? (sc_s + endc[(size_t)b * LL + e]) : neg_inf();
        srow[e] = v;
        brow[e] = s_lo | ((unsigned long long)(unsigned)e << 32);  // (s, e)
    }
}

// ---------------------------------------------------------------------------
extern "C" void kernel_launch(void* const* d_in, const int* in_sizes, int n_in,
                              void* d_out, int out_size, void* d_ws, size_t ws_size,
                              hipStream_t stream) {
    const float* text = (const float*)d_in[0];   // (B, L, D) f32
    const int*   mask = (const int*)  d_in[1];   // (B, L)    i32
    const int*   tmap = (const int*)  d_in[2];   // (B, T, 2) i32
    const float* Wg   = (const float*)d_in[3];   // (D, 3)    f32
    const float* bias = (const float*)d_in[4];   // (3,)      f32

    // workspace layout (floats)
    float* ws       = (float*)d_ws;
    float* logits   = ws;                           // B*L*3
    float* startc   = logits + (size_t)BB * LL * 3; // B*L
    float* endc     = startc + (size_t)BB * LL;     // B*L

    // outputs: scores (B*KOUT f32) then bounds (B*KOUT*2 i32, raw bits)
    float* out_scores = (float*)d_out;
    int*   out_bounds = (int*)d_out + (size_t)BB * KOUT;

    // 1) projection GEMM via f32 WMMA: 1024 row-tiles, 8 waves/block
    mention_logits_wmma<<<dim3(128), dim3(256), 0, stream>>>(text, Wg, logits);

    // 2) per-batch fold (mask, cumsum, scatter masks)
    mention_fold<<<dim3(BB), dim3(LL), 0, stream>>>(logits, mask, tmap, bias,
                                                    startc, endc);

    // 3) emit kept pairs + bounds
    mention_emit<<<dim3(LL, BB), dim3(256), 0, stream>>>(startc, endc,
                                                         out_scores, out_bounds);
}